// KernelModelSummary_17875653886470
// MI455X (gfx1250) — compile-verified
//
#include <hip/hip_runtime.h>
#include <stdint.h>

// Problem geometry (from reference): B=4096 queries, K=64 neighbors, L=256.
#define KNN   64
#define LDIM  256
#define WAVES 8        // 8 waves (wave32) per block, one query per wave
#define PFD   4        // prefetch distance in rows
#define SLOTS 8        // LDS ring slots per wave (must be > PFD, power of 2)

// 32-bit LDS byte offset for async-to-LDS VDST operand.
__device__ __forceinline__ uint32_t lds_off_u32(const void* p) {
    return (uint32_t)(size_t)(__attribute__((address_space(3))) const void*)p;
}

// Issue one gathered 1KB row (LDIM floats) into an LDS ring slot:
// 2x GLOBAL_LOAD_ASYNC_TO_LDS_B128, 32 lanes x 16B each (inst offset applies
// to BOTH global and LDS addresses per CDNA5 ISA 08 §4.4).
__device__ __forceinline__ void async_row_load(const float* __restrict__ exem,
                                               uint32_t rowByteOff,
                                               void* slot, int lane) {
    uint32_t l0    = lds_off_u32(slot) + (uint32_t)(lane * 16);
    const char* g0 = (const char*)exem + (size_t)rowByteOff + (size_t)(lane * 16);
    asm volatile("global_load_async_to_lds_b128 %0, %1, off"
                 :: "v"(l0), "v"(g0) : "memory");
    asm volatile("global_load_async_to_lds_b128 %0, %1, off offset:512"
                 :: "v"(l0), "v"(g0) : "memory");
}

__device__ __forceinline__ void fma4(float4& a, float w, const float4& r) {
    a.x = fmaf(w, r.x, a.x);
    a.y = fmaf(w, r.y, a.y);
    a.z = fmaf(w, r.z, a.z);
    a.w = fmaf(w, r.w, a.w);
}

__global__ void __launch_bounds__(WAVES * 32, 1)
knn_gather_reduce_kernel(const float* __restrict__ sq,
                         const int*   __restrict__ labels,
                         const float* __restrict__ exem,
                         const float* __restrict__ sizes,
                         const float* __restrict__ avg,
                         const float* __restrict__ tau2p,
                         const float* __restrict__ gammap,
                         float* __restrict__ out, int B) {
    __shared__ __align__(16) float s_ring[WAVES][SLOTS][LDIM];  // 64 KB
    __shared__ float    s_w[WAVES][KNN];                        // 2 KB
    __shared__ uint32_t s_off[WAVES][KNN];                      // 2 KB

    const int lane = threadIdx.x & 31;
    const int wv   = threadIdx.x >> 5;
    const int b    = blockIdx.x * WAVES + wv;
    if (b >= B) return;   // safe: no cross-wave barriers anywhere

    const float tau2  = *tau2p;
    const float gamma = *gammap;

    // ---- Phase 1: weights, row byte-offsets, denominator (k=lane, k=lane+32)
    const float d0 = sq[b * KNN + lane];
    const float d1 = sq[b * KNN + lane + 32];
    const int   i0 = labels[b * KNN + lane];
    const int   i1 = labels[b * KNN + lane + 32];
    const float w0 = (d0 <= tau2) ? __expf(-d0) : 0.0f;
    const float w1 = (d1 <= tau2) ? __expf(-d1) : 0.0f;

    float part = w0 * sizes[i0] + w1 * sizes[i1];
    s_w[wv][lane]        = w0;
    s_w[wv][lane + 32]   = w1;
    s_off[wv][lane]      = (uint32_t)i0 * (LDIM * 4u);
    s_off[wv][lane + 32] = (uint32_t)i1 * (LDIM * 4u);

#pragma unroll
    for (int off = 16; off > 0; off >>= 1)
        part += __shfl_xor(part, off, 32);
    const float rdenom = 1.0f / (part + gamma);

    // ---- Phase 2: async-DMA pipelined gather + weighted accumulate
    // Lane covers l = lane*4 .. lane*4+3  and  l = 128 + lane*4 .. +3.
    float4 acc0 = make_float4(0.f, 0.f, 0.f, 0.f);
    float4 acc1 = make_float4(0.f, 0.f, 0.f, 0.f);

#pragma unroll
    for (int k = 0; k < PFD; ++k)
        async_row_load(exem, s_off[wv][k], &s_ring[wv][k][0], lane);

    for (int k = 0; k < KNN - PFD; ++k) {
        // rows k..k+PFD-1 outstanding (2 instrs each, in-order completion):
        // <=6 outstanding  =>  row k fully landed in LDS.
        asm volatile("s_wait_asynccnt 0x6" ::: "memory");
        const float  wk   = s_w[wv][k];
        const float4* row = (const float4*)&s_ring[wv][k & (SLOTS - 1)][0];
        fma4(acc0, wk, row[lane]);
        fma4(acc1, wk, row[lane + 32]);
        // refill slot (k+PFD)&7: last read PFD iterations ago -> no WAR hazard
        async_row_load(exem, s_off[wv][k + PFD],
                       &s_ring[wv][(k + PFD) & (SLOTS - 1)][0], lane);
    }
    asm volatile("s_wait_asynccnt 0x0" ::: "memory");
    for (int k = KNN - PFD; k < KNN; ++k) {
        const float  wk   = s_w[wv][k];
        const float4* row = (const float4*)&s_ring[wv][k & (SLOTS - 1)][0];
        fma4(acc0, wk, row[lane]);
        fma4(acc1, wk, row[lane + 32]);
    }

    // ---- Epilogue: (numer + gamma*avg) / denom, 2x b128 coalesced stores
    const float4* avg4 = (const float4*)avg;
    float4* out4 = (float4*)(out + (size_t)b * LDIM);

    float4 a0 = avg4[lane];
    float4 a1 = avg4[lane + 32];
    float4 r0, r1;
    r0.x = fmaf(gamma, a0.x, acc0.x) * rdenom;
    r0.y = fmaf(gamma, a0.y, acc0.y) * rdenom;
    r0.z = fmaf(gamma, a0.z, acc0.z) * rdenom;
    r0.w = fmaf(gamma, a0.w, acc0.w) * rdenom;
    r1.x = fmaf(gamma, a1.x, acc1.x) * rdenom;
    r1.y = fmaf(gamma, a1.y, acc1.y) * rdenom;
    r1.z = fmaf(gamma, a1.z, acc1.z) * rdenom;
    r1.w = fmaf(gamma, a1.w, acc1.w) * rdenom;
    out4[lane]      = r0;
    out4[lane + 32] = r1;
}

extern "C" void kernel_launch(void* const* d_in, const int* in_sizes, int n_in,
                              void* d_out, int out_size, void* d_ws, size_t ws_size,
                              hipStream_t stream) {
    const float* sq     = (const float*)d_in[0];  // [B,K]
    const int*   labels = (const int*)  d_in[1];  // [B,K]
    const float* exem   = (const float*)d_in[2];  // [N,L]
    const float* sizes  = (const float*)d_in[3];  // [N]
    const float* avg    = (const float*)d_in[4];  // [L]
    const float* tau2   = (const float*)d_in[5];  // scalar
    const float* gamma  = (const float*)d_in[6];  // scalar
    float* out = (float*)d_out;

    const int B    = out_size / LDIM;
    const int grid = (B + WAVES - 1) / WAVES;
    hipLaunchKernelGGL(knn_gather_reduce_kernel, dim3(grid), dim3(WAVES * 32), 0,
                       stream, sq, labels, exem, sizes, avg, tau2, gamma, out, B);
    (void)in_sizes; (void)n_in; (void)d_ws; (void)ws_size;
}